// P2G_29798483099807
// MI455X (gfx1250) — compile-verified
//
#include <hip/hip_runtime.h>

namespace {
constexpr int   kGrid    = 64;
constexpr int   kG3      = kGrid * kGrid * kGrid;   // 262144 cells per batch
constexpr float kInvDx   = 64.0f;                   // 1 / DX, DX = 0.015625
constexpr float kEpsClip = 1e-5f;
constexpr float kEpsDiv  = 1e-7f;
constexpr float kHi      = 1.0f - 1e-5f;            // DX*GRID_NUM - EPS_CLIP
constexpr float kScale   = 67108864.0f;             // 2^26 fixed-point scale
constexpr float kInvScale = 1.0f / 67108864.0f;     // 2^-26
}

struct __align__(4) F3 { float x, y, z; };

// One no-return u64 atomic accumulates BOTH quantities for a cell:
//   low  32 bits: weight       * 2^26 (fixed point)
//   high 32 bits: weight*prob  * 2^26 (fixed point)
// wp <= w and per-cell weight sums are << 64, so the low half can never
// carry into the high half.
__device__ __forceinline__ void gadd_pair(unsigned long long* p, float w, float wp) {
  unsigned int wu  = __float2uint_rn(w  * kScale);
  unsigned int wpu = __float2uint_rn(wp * kScale);
  unsigned long long pk = ((unsigned long long)wpu << 32) | (unsigned long long)wu;
  (void)__hip_atomic_fetch_add(p, pk, __ATOMIC_RELAXED, __HIP_MEMORY_SCOPE_AGENT);
}

__global__ __launch_bounds__(256) void zero_kernel(float4* __restrict__ ws, int n4) {
  int i = blockIdx.x * blockDim.x + threadIdx.x;
  if (i < n4) ws[i] = float4{0.0f, 0.0f, 0.0f, 0.0f};
}

// One thread per particle: splat packed (w, w*prob) onto 27 cells.
// acc layout: one u64 per cell, batch-major: acc[b*G3 + cell]
__global__ __launch_bounds__(256) void p2g_scatter_kernel(
    const float* __restrict__ pos, const float* __restrict__ prob,
    unsigned long long* __restrict__ acc, int npart, int ntot) {
  int p = blockIdx.x * blockDim.x + threadIdx.x;
  if (p >= ntot) return;
  int b = (int)((unsigned)p / (unsigned)npart);

  F3 q = ((const F3*)pos)[p];                       // single b96 load
  float x = fminf(fmaxf(q.x, kEpsClip), kHi);
  float y = fminf(fmaxf(q.y, kEpsClip), kHi);
  float z = fminf(fmaxf(q.z, kEpsClip), kHi);

  float X = x * kInvDx, Y = y * kInvDx, Z = z * kInvDx;
  int bx = (int)X, by = (int)Y, bz = (int)Z;        // X,Y,Z > 0 => trunc == floor
  float fx = X - (float)bx, fy = Y - (float)by, fz = Z - (float)bz;

  float wX[3] = {0.5f * (1.0f - fx) * (1.0f - fx),
                 0.75f - (fx - 0.5f) * (fx - 0.5f),
                 0.5f * fx * fx};
  float wY[3] = {0.5f * (1.0f - fy) * (1.0f - fy),
                 0.75f - (fy - 0.5f) * (fy - 0.5f),
                 0.5f * fy * fy};
  float wZ[3] = {0.5f * (1.0f - fz) * (1.0f - fz),
                 0.75f - (fz - 0.5f) * (fz - 0.5f),
                 0.5f * fz * fz};

  float pr = prob[p];
  unsigned long long* a = acc + (size_t)b * (size_t)kG3;

#pragma unroll
  for (int i = 0; i < 3; ++i) {
    int tx = bx - 1 + i;
    bool okx = (unsigned)tx < (unsigned)kGrid;
    float wx = wX[i];
#pragma unroll
    for (int j = 0; j < 3; ++j) {
      int ty = by - 1 + j;
      bool okxy = okx && ((unsigned)ty < (unsigned)kGrid);
      float wxy = wx * wY[j];
      int rowbase = ((tx << 6) + ty) << 6;          // (tx*64 + ty)*64
#pragma unroll
      for (int k = 0; k < 3; ++k) {
        int tz = bz - 1 + k;
        bool ok = okxy && ((unsigned)tz < (unsigned)kGrid);
        if (ok) {
          float w = wxy * wZ[k];
          gadd_pair(a + (rowbase + tz), w, w * pr);
        }
      }
    }
  }
}

__global__ __launch_bounds__(256) void finalize_kernel(
    const uint2* __restrict__ acc, float* __restrict__ out, int n) {
  int i = blockIdx.x * blockDim.x + threadIdx.x;
  if (i < n) {
    uint2 v = acc[i];                 // v.x = w_fixed, v.y = wp_fixed
    float w  = (float)v.x * kInvScale;
    float wp = (float)v.y * kInvScale;
    out[i] = wp / (w + kEpsDiv);
  }
}

extern "C" void kernel_launch(void* const* d_in, const int* in_sizes, int n_in,
                              void* d_out, int out_size, void* d_ws, size_t ws_size,
                              hipStream_t stream) {
  const float* pos  = (const float*)d_in[0];        // (B, N, 3) f32
  const float* prob = (const float*)d_in[1];        // (B, N)    f32
  float* out = (float*)d_out;                       // (B, G3)   f32
  unsigned long long* ws = (unsigned long long*)d_ws;  // one u64 accumulator per cell

  const int ntot   = in_sizes[1];                   // B*N particles (2,097,152)
  const int nbatch = out_size / kG3;                // 16
  const int npart  = ntot / (nbatch > 0 ? nbatch : 1);  // 131072

  const int accDwords = 2 * out_size;               // 8,388,608 dwords (33.5 MB) in ws
  const int n4 = accDwords / 4;

  zero_kernel<<<(n4 + 255) / 256, 256, 0, stream>>>((float4*)d_ws, n4);
  p2g_scatter_kernel<<<(ntot + 255) / 256, 256, 0, stream>>>(pos, prob, ws, npart, ntot);
  finalize_kernel<<<(out_size + 255) / 256, 256, 0, stream>>>((const uint2*)d_ws, out, out_size);
}